// OutputPhysNet_2774548873947
// MI455X (gfx1250) — compile-verified
//
#include <hip/hip_runtime.h>
#include <hip/hip_bf16.h>
#include <stdint.h>

#define BB   16
#define AA   1024
#define AM1  1023
#define KE_F 0.62415089925f

// d_out float layout (flat, return order):
//  E:[0,16)  Q:[16,32)  q_eff:[32,16416)  dipole:[16416,16464)  Ei_masked:[16464,32848)
#define OFF_Q    16
#define OFF_QEFF 32
#define OFF_DIP  16416
#define OFF_EIM  16464

typedef __attribute__((ext_vector_type(2))) float v2f;
typedef __attribute__((ext_vector_type(8))) float v8f;
typedef __attribute__((ext_vector_type(4))) int   v4i;

// wave32 xor-swizzle reduction: ds_swizzle group-of-32, imm = (xor_mask<<10)|0x1F
__device__ __forceinline__ float wave_red32(float v) {
  v += __int_as_float(__builtin_amdgcn_ds_swizzle(__float_as_int(v), 0x401F)); // x16
  v += __int_as_float(__builtin_amdgcn_ds_swizzle(__float_as_int(v), 0x201F)); // x8
  v += __int_as_float(__builtin_amdgcn_ds_swizzle(__float_as_int(v), 0x101F)); // x4
  v += __int_as_float(__builtin_amdgcn_ds_swizzle(__float_as_int(v), 0x081F)); // x2
  v += __int_as_float(__builtin_amdgcn_ds_swizzle(__float_as_int(v), 0x041F)); // x1
  return v;
}
// sum within each 16-lane half (for WMMA row extraction)
__device__ __forceinline__ float half_red16(float v) {
  v += __int_as_float(__builtin_amdgcn_ds_swizzle(__float_as_int(v), 0x201F)); // x8
  v += __int_as_float(__builtin_amdgcn_ds_swizzle(__float_as_int(v), 0x101F)); // x4
  v += __int_as_float(__builtin_amdgcn_ds_swizzle(__float_as_int(v), 0x081F)); // x2
  v += __int_as_float(__builtin_amdgcn_ds_swizzle(__float_as_int(v), 0x041F)); // x1
  return v;
}

// coulomb_approx: fc = physnet_cutoff(2r, rc=5) = 1 - 6u^5 + 15u^4 - 10u^3, u = 2r/5
// result = fc/sqrt(r^2+1) + (1-fc)/(r+1e-6)
__device__ __forceinline__ float coulomb_approx(float r) {
  float u     = r * 0.4f;
  float u2    = u * u;
  float inner = fmaf(u, fmaf(u, 6.0f, -15.0f), 10.0f);   // 10 - 15u + 6u^2
  float f     = fmaf(-(u2 * u), inner, 1.0f);            // 1 - u^3*inner
  float fc    = (r < 2.5f) ? f : 0.0f;
  float rs    = __builtin_amdgcn_rsqf(fmaf(r, r, 1.0f)); // v_rsq_f32
  float rc    = __builtin_amdgcn_rcpf(r + 1e-6f);        // v_rcp_f32
  return fmaf(fc, rs, (1.0f - fc) * rc);
}

// ---------------- Kernel 1: per-batch sums, q_eff, dipole, Ei_masked ----------------
__global__ __launch_bounds__(1024)
void k1_per_atom(const float* __restrict__ atom_mask,
                 const float* __restrict__ energies,
                 const float* __restrict__ charges,
                 const float* __restrict__ positions,
                 float* __restrict__ out,
                 float* __restrict__ Ebase_ws) {
  __shared__ float sred[3][32];
  __shared__ float sbr[3];
  const int b = blockIdx.x, t = threadIdx.x;
  const int wid = t >> 5;
  const int idx = b * AA + t;

  const float m  = atom_mask[idx];
  const float e  = energies[idx];
  const float q  = charges[idx];
  const float Ei = fmaf(e, 1.0f /*STDDEV*/, 0.0f /*MEAN*/);
  const float vE = Ei * m;
  const float vQ = q * m;

  float rE = wave_red32(vE), rQ = wave_red32(vQ), rN = wave_red32(m);
  if ((t & 31) == 0) { sred[0][wid] = rE; sred[1][wid] = rQ; sred[2][wid] = rN; }
  __syncthreads();
  if (t < 32) {
    float xE = wave_red32(sred[0][t]);
    float xQ = wave_red32(sred[1][t]);
    float xN = wave_red32(sred[2][t]);
    if (t == 0) { sbr[0] = xE; sbr[1] = xQ; sbr[2] = xN; }
  }
  __syncthreads();

  const float Qs = sbr[1], Ns = sbr[2];
  const float qeff = (vQ - Qs / Ns) * m;
  out[OFF_QEFF + idx] = qeff;
  out[OFF_EIM  + idx] = e * m;

  const float px = positions[idx * 3 + 0];
  const float py = positions[idx * 3 + 1];
  const float pz = positions[idx * 3 + 2];
  float dx = wave_red32(px * qeff);
  float dy = wave_red32(py * qeff);
  float dz = wave_red32(pz * qeff);
  __syncthreads();
  if ((t & 31) == 0) { sred[0][wid] = dx; sred[1][wid] = dy; sred[2][wid] = dz; }
  __syncthreads();
  if (t < 32) {
    float rx = wave_red32(sred[0][t]);
    float ry = wave_red32(sred[1][t]);
    float rz = wave_red32(sred[2][t]);
    if (t == 0) {
      out[OFF_Q + b]           = sbr[1];
      Ebase_ws[b]              = sbr[0];
      out[OFF_DIP + b * 3 + 0] = rx;
      out[OFF_DIP + b * 3 + 1] = ry;
      out[OFF_DIP + b * 3 + 2] = rz;
    }
  }
}

// ---------------- Kernel 2: streaming Coulomb reduction (one wave per row) ----------------
// All 8 rows of a block share one batch -> stage q_eff row (4KB) in LDS once,
// via CDNA5 async global->LDS if the builtin is available.
__global__ __launch_bounds__(256)
void k2_coulomb(const float* __restrict__ dist,
                const float* __restrict__ qe,     // = out + OFF_QEFF, (B,A)
                float* __restrict__ partials) {
  __shared__ __align__(16) float sqe[AA];
  __shared__ float wsum[8];
  const int t    = threadIdx.x;
  const int wid  = t >> 5;
  const int lane = t & 31;
  const int row0 = blockIdx.x * 8;                // rows [row0, row0+8) in one batch
  const int b    = row0 >> 10;
  const float* __restrict__ qrow = qe + b * AA;

  // cooperative stage: 256 threads x 16B = 4KB
#if __has_builtin(__builtin_amdgcn_global_load_async_to_lds_b128)
  __builtin_amdgcn_global_load_async_to_lds_b128(
      (__attribute__((address_space(1))) v4i*)(uintptr_t)(qrow + t * 4),
      (__attribute__((address_space(3))) v4i*)(uintptr_t)(&sqe[t * 4]),
      0, 0);
  asm volatile("s_wait_asynccnt 0x0" ::: "memory");
#else
  *(float4*)(&sqe[t * 4]) = *(const float4*)(qrow + t * 4);
#endif
  __syncthreads();

  const int row = row0 + wid;
  const int i   = row & (AA - 1);
  const float* __restrict__ drow = dist + (size_t)row * AM1;
  __builtin_prefetch(drow + lane, 0, 3);          // global_prefetch_b8

  float acc = 0.0f;
  // k < i : j = k   (no per-element diagonal compare)
  for (int k = lane; k < i; k += 32)
    acc = fmaf(coulomb_approx(drow[k]), sqe[k], acc);
  // k >= i : j = k+1 (constant +4B folds into the ds_load offset)
  for (int k = i + lane; k < AM1; k += 32)
    acc = fmaf(coulomb_approx(drow[k]), sqe[k + 1], acc);
  acc *= sqe[i];

  float s = wave_red32(acc);
  if (lane == 0) wsum[wid] = s;
  __syncthreads();
  if (t == 0) {
    float tot = 0.0f;
#pragma unroll
    for (int w = 0; w < 8; ++w) tot += wsum[w];   // fixed order -> deterministic
    partials[blockIdx.x] = tot;
  }
}

// ---------------- Kernel 3: WMMA f32 reduction of 2048 partials + finalize E ----------------
// D = ones(16x4) x B(4x16) + C  accumulates column sums of 64 f32 values per WMMA.
__global__ __launch_bounds__(128)
void k3_finalize(const float* __restrict__ partials,
                 const float* __restrict__ Ebase,
                 float* __restrict__ out) {
  __shared__ float wtot[4];
  __shared__ float scal;
  const int t = threadIdx.x;
  const int wid = t >> 5, lane = t & 31;
  const float* __restrict__ base = partials + wid * 512;   // 512 partials per wave

  v2f a; a[0] = 1.0f; a[1] = 1.0f;      // A = all-ones 16x4
  v8f c = {};
#pragma unroll
  for (int it = 0; it < 8; ++it) {      // 8 x 64 = 512 values folded into C
    v2f bv = *(const v2f*)(base + it * 64 + lane * 2);
    c = __builtin_amdgcn_wmma_f32_16x16x4_f32(
        /*neg_a=*/false, a, /*neg_b=*/false, bv,
        /*c_mod=*/(short)0, c, /*reuse_a=*/false, /*reuse_b=*/false);
  }
  // D[m,n] = colsum[n] for every m; lanes 0..15 of c[0] hold row M=0 (N=0..15)
  float v = half_red16(c[0]);
  if (lane == 0) wtot[wid] = v;
  __syncthreads();
  if (t == 0) scal = 0.5f * KE_F * (wtot[0] + wtot[1] + wtot[2] + wtot[3]);
  __syncthreads();
  if (t < BB) out[t] = Ebase[t] + scal;  // E = E_base + K_E/2 * coulomb_sum
}

extern "C" void kernel_launch(void* const* d_in, const int* in_sizes, int n_in,
                              void* d_out, int out_size, void* d_ws, size_t ws_size,
                              hipStream_t stream) {
  // inputs: 0 atomic_numbers(i32, unused) 1 positions 2 atom_mask 3 energies 4 charges 5 distances
  const float* positions = (const float*)d_in[1];
  const float* atom_mask = (const float*)d_in[2];
  const float* energies  = (const float*)d_in[3];
  const float* charges   = (const float*)d_in[4];
  const float* distances = (const float*)d_in[5];
  float* out = (float*)d_out;
  float* ws  = (float*)d_ws;
  float* Ebase    = ws;        // 16 floats
  float* partials = ws + 16;   // 2048 floats

  hipLaunchKernelGGL(k1_per_atom, dim3(BB), dim3(1024), 0, stream,
                     atom_mask, energies, charges, positions, out, Ebase);
  hipLaunchKernelGGL(k2_coulomb, dim3(2048), dim3(256), 0, stream,
                     distances, out + OFF_QEFF, partials);
  hipLaunchKernelGGL(k3_finalize, dim3(1), dim3(128), 0, stream,
                     partials, Ebase, out);
}